// Lin2d_39470749450322
// MI455X (gfx1250) — compile-verified
//
#include <hip/hip_runtime.h>
#include <cmath>

// CDNA5 / gfx1250: wave32, WMMA 16x16x4 f32.
typedef __attribute__((ext_vector_type(2))) float v2f;
typedef __attribute__((ext_vector_type(4))) float v4f;
typedef __attribute__((ext_vector_type(8))) float v8f;

#define WAVES_PER_BLOCK 8
#define TILES_PER_WAVE  8
#define FLOATS_PER_TILE 64   // 32 states * 2 floats = one WMMA's worth of B data
#define CHUNK_FLOATS    (TILES_PER_WAVE * FLOATS_PER_TILE)

// D = A(16x4) * B(4x16): A carries the 2x2 rotation twice (block-diagonal in
// rows 0..3), B carries 32 states (two per column). Per the 32-bit WMMA
// layouts, lane L / component j holds:
//   A[L%16][2*(L>>4)+j]   and   B[2*(L>>4)+j][L%16]
// so B is a contiguous float2 per lane (coalesced 256B per wave-tile), and the
// result rows 0..3 sit in lanes 0..15 VGPRs 0..3 = contiguous float4 stores.
__global__ __launch_bounds__(256) void
lin2d_rot_wmma(const float* __restrict__ in, float* __restrict__ out,
               float c, float s, long long nFloats) {
    const int lane = threadIdx.x & 31;
    const int wave = threadIdx.x >> 5;
    const int m    = lane & 15;   // row (A) / column (B) index
    const int half = lane >> 4;   // selects K in {0,1} vs K in {2,3}

    // Build the constant A matrix fragment for this lane.
    v2f a; a.x = 0.0f; a.y = 0.0f;
    if (half == 0) {                       // K = 0,1
        if (m == 0)      { a.x =  c; a.y = s; }   // A[0][0..1] = [ c, s]
        else if (m == 1) { a.x = -s; a.y = c; }   // A[1][0..1] = [-s, c]
    } else {                               // K = 2,3
        if (m == 2)      { a.x =  c; a.y = s; }   // A[2][2..3] = [ c, s]
        else if (m == 3) { a.x = -s; a.y = c; }   // A[3][2..3] = [-s, c]
    }

    const long long waveBase =
        ((long long)blockIdx.x * WAVES_PER_BLOCK + wave) * (long long)CHUNK_FLOATS;
    const int loadOff = 4 * m + 2 * half;  // float2 per lane, contiguous per wave

    if (waveBase + CHUNK_FLOATS <= nFloats) {
        // ---- Fast path: whole chunk in range; no per-tile branches. ----
        // Phase 1: issue all loads (8 outstanding b64 loads per wave).
        v2f b[TILES_PER_WAVE];
#pragma unroll
        for (int t = 0; t < TILES_PER_WAVE; ++t) {
            b[t] = __builtin_nontemporal_load(
                (const v2f*)(in + waveBase + (long long)t * FLOATS_PER_TILE + loadOff));
        }
        // Phase 2: all WMMAs with full EXEC (ISA requirement).
        v4f o[TILES_PER_WAVE];
#pragma unroll
        for (int t = 0; t < TILES_PER_WAVE; ++t) {
            v8f acc = {};
            // 8 args: (neg_a, A, neg_b, B, c_mod, C, reuse_a, reuse_b)
            v8f d = __builtin_amdgcn_wmma_f32_16x16x4_f32(
                false, a, false, b[t], (short)0, acc, false, false);
            o[t].x = d[0]; o[t].y = d[1]; o[t].z = d[2]; o[t].w = d[3];
        }
        // Phase 3: single divergent region, 8 coalesced b128 stores.
        if (half == 0) {
#pragma unroll
            for (int t = 0; t < TILES_PER_WAVE; ++t) {
                __builtin_nontemporal_store(
                    o[t],
                    (v4f*)(out + waveBase + (long long)t * FLOATS_PER_TILE + 4 * m));
            }
        }
    } else {
        // ---- Tail path: per-tile uniform guard (never taken for exact grids). ----
        for (int t = 0; t < TILES_PER_WAVE; ++t) {
            const long long tb = waveBase + (long long)t * FLOATS_PER_TILE;
            if (tb + FLOATS_PER_TILE > nFloats) break;  // wave-uniform
            v2f b = *(const v2f*)(in + tb + loadOff);
            v8f acc = {};
            v8f d = __builtin_amdgcn_wmma_f32_16x16x4_f32(
                false, a, false, b, (short)0, acc, false, false);
            if (half == 0) {
                v4f o; o.x = d[0]; o.y = d[1]; o.z = d[2]; o.w = d[3];
                *(v4f*)(out + tb + 4 * m) = o;
            }
        }
    }
}

extern "C" void kernel_launch(void* const* d_in, const int* in_sizes, int n_in,
                              void* d_out, int out_size, void* d_ws, size_t ws_size,
                              hipStream_t stream) {
    (void)in_sizes; (void)n_in; (void)d_ws; (void)ws_size;
    const float* x = (const float*)d_in[0];
    float* out = (float*)d_out;

    // Collapse 1000 rotations of pi/100 into a single rotation (exact in
    // exact arithmetic; deviates from the fp32 step-loop by O(N*eps)).
    const double theta = 3.14159265358979323846 / 100.0;
    const double ang = 1000.0 * theta;
    const float c = (float)std::cos(ang);
    const float s = (float)std::sin(ang);

    const long long nFloats = (long long)out_size;  // BATCH * 2
    const long long floatsPerBlock = (long long)WAVES_PER_BLOCK * CHUNK_FLOATS;
    const int grid = (int)((nFloats + floatsPerBlock - 1) / floatsPerBlock);

    lin2d_rot_wmma<<<grid, 256, 0, stream>>>(x, out, c, s, nFloats);
}